// BoxListSoftNMS_49658411876612
// MI455X (gfx1250) — compile-verified
//
#include <hip/hip_runtime.h>

// Soft-NMS, exact sequential semantics (B=64 batches, N=4096 boxes).
// One workgroup per batch; 256 threads (8 wave32s), 16 boxes per thread in VGPRs.
// CDNA5 paths: global_load_async_to_lds_b128 staging + s_wait_asynccnt,
// wave32 shuffle argmax, single s_barrier per scan step (double-buffered LDS slots).
// Frozen boxes are encoded as ws = -inf (no mask ops in argmax/decay); final
// score + keep flag are stored by the owning thread at freeze time.

#define NBOX 4096
#define TPB  256
#define KPT  (NBOX / TPB)   // 16 boxes per thread
#define NWAV (TPB / 32)     // 8 waves

__global__ __launch_bounds__(TPB)
void BoxListSoftNMS_kernel(const float* __restrict__ boxes,
                           const float* __restrict__ scores,
                           float* __restrict__ out)
{
    __shared__ float4 sbox[NBOX];        // 64 KB: staged box table for this batch
    __shared__ float2 red[2][NWAV];      // double-buffered (value, index) argmax slots

    const int b = blockIdx.x;
    const int t = threadIdx.x;
    const float4* __restrict__ gbox = (const float4*)boxes + (size_t)b * NBOX;

    // ---- stage this batch's boxes into LDS via CDNA5 async global->LDS DMA ----
#pragma unroll
    for (int k = 0; k < KPT; ++k) {
        int g = (k << 8) | t;                                // k*256 + t (coalesced)
        unsigned lds_off = (unsigned)(size_t)(&sbox[g]);     // low 32 bits = LDS offset
        unsigned long long gaddr = (unsigned long long)(size_t)(gbox + g);
        asm volatile("global_load_async_to_lds_b128 %0, %1, off"
                     :: "v"(lds_off), "v"(gaddr) : "memory");
    }
    asm volatile("s_wait_asynccnt 0" ::: "memory");
    __syncthreads();

    // ---- per-thread register-resident state ----
    float x1[KPT], y1[KPT], x2[KPT], y2[KPT], ar[KPT], ws[KPT];
#pragma unroll
    for (int k = 0; k < KPT; ++k) {
        int g = (k << 8) | t;
        float4 bx = sbox[g];
        x1[k] = bx.x; y1[k] = bx.y; x2[k] = bx.z; y2[k] = bx.w;
        ar[k] = (bx.z - bx.x) * (bx.w - bx.y);   // matches reference areas
        ws[k] = scores[(size_t)b * NBOX + g];    // working score; -inf once frozen
    }

    const float  NEG   = -__builtin_inff();
    const size_t obase = (size_t)b * NBOX;
    const size_t total = (size_t)gridDim.x * NBOX;

    // ---- sequential Soft-NMS scan: N dependent steps ----
    for (int step = 0; step < NBOX; ++step) {
        // local argmax: frozen are -inf, never selected while live boxes remain.
        // ascending global index + strict '>' == jnp.argmax first-max tie-break.
        float bv = NEG; int bi = 0;
#pragma unroll
        for (int k = 0; k < KPT; ++k) {
            float v = ws[k];
            int gi = (k << 8) | t;
            bool take = v > bv;
            bv = take ? v : bv;
            bi = take ? gi : bi;
        }
        // wave32 butterfly reduce (smallest index wins ties)
#pragma unroll
        for (int off = 16; off > 0; off >>= 1) {
            float ov = __shfl_xor(bv, off, 32);
            int   oi = __shfl_xor(bi, off, 32);
            bool take = (ov > bv) || (ov == bv && oi < bi);
            bv = take ? ov : bv;
            bi = take ? oi : bi;
        }
        // cross-wave combine: double-buffered slots => exactly one barrier per step
        const int p = step & 1;
        if ((t & 31) == 0) red[p][t >> 5] = make_float2(bv, __int_as_float(bi));
        __syncthreads();
        float wv = red[p][0].x; int wi = __float_as_int(red[p][0].y);
#pragma unroll
        for (int j = 1; j < NWAV; ++j) {
            float2 r = red[p][j];
            bool take = (r.x > wv) || (r.x == wv && __float_as_int(r.y) < wi);
            wv = take ? r.x : wv;
            wi = take ? __float_as_int(r.y) : wi;
        }
        // winner box: same-address LDS broadcast read
        float4 wb = sbox[wi];
        float warea = (wb.z - wb.x) * (wb.w - wb.y);
        // owner emits the final score + keep flag at freeze time (wv is bit-exact)
        if ((wi & (TPB - 1)) == t) {
            size_t o = obase + (size_t)wi;
            out[o] = wv;
            out[total + o] = (wv >= 0.05f) ? 1.0f : 0.0f;
        }
        // decay (maskless): -inf propagates through both arms; the just-selected
        // element is forced to -inf by the compare-select (freeze-before-decay
        // semantics preserved since its output was already written from wv).
#pragma unroll
        for (int k = 0; k < KPT; ++k) {
            float iw = fminf(x2[k], wb.z) - fmaxf(x1[k], wb.x);
            float ih = fminf(y2[k], wb.w) - fmaxf(y1[k], wb.y);
            iw = fmaxf(iw, 0.0f); ih = fmaxf(ih, 0.0f);
            float inter = iw * ih;
            // association matches reference: ((areas + barea) - inter) + EPS
            float denom = ((ar[k] + warea) - inter) + 1e-6f;
            float iou = inter / denom;                         // IEEE div: bit-faithful
            float dws = (iou > 0.5f) ? ws[k] * (1.0f - iou) : ws[k];
            ws[k] = (wi == ((k << 8) | t)) ? NEG : dws;
        }
    }
}

extern "C" void kernel_launch(void* const* d_in, const int* in_sizes, int n_in,
                              void* d_out, int out_size, void* d_ws, size_t ws_size,
                              hipStream_t stream) {
    const float* boxes  = (const float*)d_in[0];   // [B, N, 4] f32
    const float* scores = (const float*)d_in[1];   // [B, N]    f32
    float* out = (float*)d_out;                    // [B*N] scores ++ [B*N] keep
    const int nbatch = in_sizes[1] / NBOX;         // B = 64
    BoxListSoftNMS_kernel<<<dim3(nbatch), dim3(TPB), 0, stream>>>(boxes, scores, out);
}